// GatedGCN_51238959841304
// MI455X (gfx1250) — compile-verified
//
#include <hip/hip_runtime.h>
#include <math.h>

// GatedGCN forward for MI455X (gfx1250, wave32, WMMA).
// D = 128 fixed. fp32 WMMA 16x16x4 for all GEMMs (problem is L2/atomic bound;
// f32 keeps reference precision). Edge aggregation via hardware
// global_atomic_add_f32. Self-loop scaling fused into the GEMM epilogue.

#define DF 128
#define LDSS 132  // padded LDS row stride (floats); 132 % 64 == 4 -> conflict-free b64 reads

typedef __attribute__((ext_vector_type(2))) float v2f;
typedef __attribute__((ext_vector_type(8))) float v8f;

// ------------------------- elementwise kernels -------------------------

// y = clip(x, -100, 100), float4-vectorized (n4 = n/4 elements of float4)
__global__ void k_clip4(const float4* __restrict__ x, float4* __restrict__ y, int n4) {
  int i = blockIdx.x * 256 + threadIdx.x;
  if (i < n4) {
    float4 v = x[i];
    v.x = fminf(fmaxf(v.x, -100.f), 100.f);
    v.y = fminf(fmaxf(v.y, -100.f), 100.f);
    v.z = fminf(fmaxf(v.z, -100.f), 100.f);
    v.w = fminf(fmaxf(v.w, -100.f), 100.f);
    y[i] = v;
  }
}

__global__ void k_deg_init(int* __restrict__ deg, int n) {
  int i = blockIdx.x * 256 + threadIdx.x;
  if (i < n) deg[i] = 1;  // self-loop contributes 1 to in-degree
}

__global__ void k_deg_count(const int* __restrict__ dst, int* __restrict__ deg, int nE) {
  int e = blockIdx.x * 256 + threadIdx.x;
  if (e < nE) atomicAdd(&deg[dst[e]], 1);
}

__global__ void k_deg_rsqrt(float* __restrict__ dinv, int n) {
  int i = blockIdx.x * 256 + threadIdx.x;
  if (i < n) {
    int d = __float_as_int(dinv[i]);  // buffer held ints during counting
    dinv[i] = rsqrtf((float)d);       // deg >= 1 always
  }
}

// B = relu(B + bias[col]) (+ res if non-null), in place, float4-vectorized
__global__ void k_bias_relu_res4(float4* __restrict__ B, const float4* __restrict__ bias,
                                 const float4* __restrict__ res, int n4) {
  int i = blockIdx.x * 256 + threadIdx.x;
  if (i < n4) {
    float4 v = B[i];
    float4 bb = bias[i & (DF / 4 - 1)];
    v.x = fmaxf(v.x + bb.x, 0.f);
    v.y = fmaxf(v.y + bb.y, 0.f);
    v.z = fmaxf(v.z + bb.z, 0.f);
    v.w = fmaxf(v.w + bb.w, 0.f);
    if (res) {
      float4 r = res[i];
      v.x += r.x; v.y += r.y; v.z += r.z; v.w += r.w;
    }
    B[i] = v;
  }
}

// ------------------------- edge scatter -------------------------
// One wave32 per edge; each lane handles 4 features: float4 gather from the
// source row, scaled, 4 hardware float atomics into the destination row.
__global__ void k_scatter(const float* __restrict__ H, const int* __restrict__ src,
                          const int* __restrict__ dst, const float* __restrict__ dinv,
                          float* __restrict__ AGG, int nE) {
  int lane = threadIdx.x & 31;
  int e = blockIdx.x * 8 + (threadIdx.x >> 5);
  if (e >= nE) return;
  int s = src[e];
  int d = dst[e];
  float c = dinv[s] * dinv[d];
  const float4 v = *(const float4*)(H + (size_t)s * DF + lane * 4);
  float* ap = AGG + (size_t)d * DF + lane * 4;
  unsafeAtomicAdd(ap + 0, v.x * c);
  unsafeAtomicAdd(ap + 1, v.y * c);
  unsafeAtomicAdd(ap + 2, v.z * c);
  unsafeAtomicAdd(ap + 3, v.w * c);
}

// ------------------------- WMMA GEMM -------------------------
// acc = X[n,128] @ W[128,128] (+ X2 @ W2 if X2 != null)
// Epilogues:
//   default:        Y = acc; and if Y2 != null:  Y2 = acc * dinv[row]^2
//                   (fused self-loop init of the aggregation buffer)
//   gate (X2!=0):   Y = mul * sigmoid(acc + bias)
// Block: 256 threads = 8 waves; block covers 128 rows; each wave 16 rows x
// 128 cols = 8 accumulator tiles of v_wmma_f32_16x16x4_f32.
// W is staged transposed into LDS (stride LDSS) so B-fragments are b64 loads.
__global__ __launch_bounds__(256) void k_gemm128(
    const float* __restrict__ X, const float* __restrict__ W,
    const float* __restrict__ X2, const float* __restrict__ W2,
    const float* __restrict__ bias, const float* __restrict__ mul,
    const float* __restrict__ dinv, float* __restrict__ Y,
    float* __restrict__ Y2, int n) {
  extern __shared__ float lds[];
  const int tid = threadIdx.x;
  const int lane = tid & 31;
  const int wave = tid >> 5;
  const int row_base = blockIdx.x * 128 + wave * 16;
  const bool active = row_base < n;  // n is a multiple of 16

  v8f acc[8];
#pragma unroll
  for (int t = 0; t < 8; ++t) acc[t] = {};

  // stage W^T into LDS: lds[col*LDSS + k] = W[k*128 + col] (coalesced global reads)
  for (int idx = tid; idx < DF * DF; idx += 256) {
    int k = idx >> 7;
    int c = idx & (DF - 1);
    lds[c * LDSS + k] = W[idx];
  }
  __syncthreads();

  const int rlane = lane & 15;  // row within 16-row tile (A) / col within tile (B,C,D)
  const int ksel = lane >> 4;   // K sub-select per ISA 16x4 f32 layout

  if (active) {
    const float* ap = X + (size_t)(row_base + rlane) * DF + 2 * ksel;
    const float* wp = lds + rlane * LDSS + 2 * ksel;
#pragma unroll 4
    for (int kk = 0; kk < 32; ++kk) {
      v2f a = *(const v2f*)(ap + 4 * kk);
#pragma unroll
      for (int t = 0; t < 8; ++t) {
        v2f b = *(const v2f*)(wp + t * 16 * LDSS + 4 * kk);
        acc[t] = __builtin_amdgcn_wmma_f32_16x16x4_f32(
            false, a, false, b, (short)0, acc[t], false, false);
      }
    }
  }

  if (X2 != nullptr) {  // uniform branch: barriers are safe
    __syncthreads();
    for (int idx = tid; idx < DF * DF; idx += 256) {
      int k = idx >> 7;
      int c = idx & (DF - 1);
      lds[c * LDSS + k] = W2[idx];
    }
    __syncthreads();
    if (active) {
      const float* ap = X2 + (size_t)(row_base + rlane) * DF + 2 * ksel;
      const float* wp = lds + rlane * LDSS + 2 * ksel;
#pragma unroll 4
      for (int kk = 0; kk < 32; ++kk) {
        v2f a = *(const v2f*)(ap + 4 * kk);
#pragma unroll
        for (int t = 0; t < 8; ++t) {
          v2f b = *(const v2f*)(wp + t * 16 * LDSS + 4 * kk);
          acc[t] = __builtin_amdgcn_wmma_f32_16x16x4_f32(
              false, a, false, b, (short)0, acc[t], false, false);
        }
      }
    }
  }

  if (!active) return;

  // C/D layout: VGPR r, lane L -> (row = r + 8*(L/16), col = L%16) of the tile
  const int col0 = lane & 15;
  const int rsel = lane >> 4;
  const bool gate = (X2 != nullptr);

  float dsc[8];
  if (Y2 != nullptr) {
#pragma unroll
    for (int r = 0; r < 8; ++r) {
      float d = dinv[row_base + r + 8 * rsel];
      dsc[r] = d * d;
    }
  }

#pragma unroll
  for (int t = 0; t < 8; ++t) {
    int col = t * 16 + col0;
    float bb = bias ? bias[col] : 0.f;
#pragma unroll
    for (int r = 0; r < 8; ++r) {
      int row = row_base + r + 8 * rsel;
      size_t off = (size_t)row * DF + col;
      float v = acc[t][r] + bb;
      if (gate) {
        float g = 1.f / (1.f + __expf(-v));
        Y[off] = mul[off] * g;
      } else {
        Y[off] = v;
        if (Y2 != nullptr) Y2[off] = v * dsc[r];
      }
    }
  }
}

// ------------------------- launch -------------------------

extern "C" void kernel_launch(void* const* d_in, const int* in_sizes, int n_in,
                              void* d_out, int out_size, void* d_ws, size_t ws_size,
                              hipStream_t stream) {
  const float* x  = (const float*)d_in[0];
  const int*   ei = (const int*)d_in[1];
  const float* W1 = (const float*)d_in[2];
  const float* b1 = (const float*)d_in[3];
  const float* W2 = (const float*)d_in[4];
  const float* b2 = (const float*)d_in[5];
  const float* Wg = (const float*)d_in[6];
  const float* bg = (const float*)d_in[7];

  const int N = in_sizes[0] / DF;
  const int E = in_sizes[1] / 2;
  const int* src = ei;
  const int* dst = ei + E;
  float* out = (float*)d_out;

  const size_t NT = (size_t)N * DF;
  float* X0   = (float*)d_ws;   // clipped input
  float* A    = X0 + NT;        // GEMM output / messages
  float* B    = A + NT;         // aggregation / h1
  float* C    = B + NT;         // aggregation2 / h
  float* dinv = C + NT;         // deg (as int) then deg^-1/2

  const int nel  = N * DF;
  const int nel4 = nel / 4;
  const dim3 blk(256);
  const int gb_el4 = (nel4 + 255) / 256;
  const int gb_n   = (N + 255) / 256;
  const int gb_e   = (E + 255) / 256;
  const int gb_sc  = (E + 7) / 8;          // one wave per edge, 8 waves/block
  const int gb_mm  = (N + 127) / 128;      // 128 rows per block
  const size_t lds_bytes = (size_t)LDSS * DF * sizeof(float);

  // degree + normalization
  k_clip4<<<gb_el4, blk, 0, stream>>>((const float4*)x, (float4*)X0, nel4);
  k_deg_init<<<gb_n, blk, 0, stream>>>((int*)dinv, N);
  k_deg_count<<<gb_e, blk, 0, stream>>>(dst, (int*)dinv, E);
  k_deg_rsqrt<<<gb_n, blk, 0, stream>>>(dinv, N);

  // conv1: A = X0 @ W1 (fused: B = A * dinv^2) ; B += scatter(A) ; B = relu(B + b1)
  k_gemm128<<<gb_mm, blk, lds_bytes, stream>>>(X0, W1, nullptr, nullptr, nullptr, nullptr,
                                               dinv, A, B, N);
  k_scatter<<<gb_sc, blk, 0, stream>>>(A, src, dst, dinv, B, E);
  k_bias_relu_res4<<<gb_el4, blk, 0, stream>>>((float4*)B, (const float4*)b1, nullptr, nel4);

  // conv2: A = B @ W2 (fused: C = A * dinv^2) ; C += scatter(A) ; C = relu(C + b2) + X0
  k_gemm128<<<gb_mm, blk, lds_bytes, stream>>>(B, W2, nullptr, nullptr, nullptr, nullptr,
                                               dinv, A, C, N);
  k_scatter<<<gb_sc, blk, 0, stream>>>(A, src, dst, dinv, C, E);
  k_bias_relu_res4<<<gb_el4, blk, 0, stream>>>((float4*)C, (const float4*)b2,
                                               (const float4*)X0, nel4);

  // gate: out = C * sigmoid(C @ Wg[0:128] + X0 @ Wg[128:256] + bg)
  k_gemm128<<<gb_mm, blk, lds_bytes, stream>>>(C, Wg, X0, Wg + DF * DF, bg, C,
                                               nullptr, out, nullptr, N);
}